// Discriminator_37967510897363
// MI455X (gfx1250) — compile-verified
//
#include <hip/hip_runtime.h>

typedef float v2f __attribute__((ext_vector_type(2)));
typedef float v8f __attribute__((ext_vector_type(8)));

#define N_NODES   200000
#define N_H       256
#define N_SEG     1024
#define SEG_SIZE  128
#define N_POS     (N_SEG * SEG_SIZE)     // 131072
#define NEG_RATIO 5
#define N_NEG     (N_POS * NEG_RATIO)    // 655360
#define NEG_PER_SEG (SEG_SIZE * NEG_RATIO) // 640

// ---------------------------------------------------------------------------
// Kernel 1: xbar[s] = mean_{j in segment s} embedding[pos_samples[s*128+j]]
// One 256-thread block per segment; each thread owns one column -> every row
// read is a fully-coalesced 1KB burst across the block.
// ---------------------------------------------------------------------------
__global__ void k_segmean(const float* __restrict__ emb,
                          const int* __restrict__ pos,
                          const int* __restrict__ gsz,
                          float* __restrict__ xbar) {
    const int s = blockIdx.x;
    const int h = threadIdx.x;
    const int base = s * SEG_SIZE;
    float acc = 0.0f;
#pragma unroll 4
    for (int j = 0; j < SEG_SIZE; ++j) {
        const int idx = pos[base + j];
        acc += emb[(size_t)idx * N_H + h];
    }
    xbar[(size_t)s * N_H + h] = acc / (float)gsz[s];
}

// ---------------------------------------------------------------------------
// Kernel 2: fp32 WMMA GEMM  C[M,N] = A[M,K] @ (TRANSB ? B^T : B) (+ bias[n])
// One wave per 16x16 C tile, V_WMMA_F32_16X16X4_F32, K stepped by 4.
// f32 A 16x4 layout: lanes 0-15 -> M rows, VGPR{0,1} = K{0,1}; lanes 16-31 =
// K{2,3}.  B 4x16 symmetric (N on lanes, halves split K).  EXEC all-1s:
// 32-thread blocks, no divergence, M/N/K all multiples of 16/4.
// ---------------------------------------------------------------------------
template <int TRANSB, int BIAS>
__global__ void k_gemm_wmma(const float* __restrict__ A,
                            const float* __restrict__ B,
                            const float* __restrict__ bias,
                            float* __restrict__ C,
                            int M, int N, int K) {
    const int lane = threadIdx.x;          // 0..31
    const int half = lane >> 4;            // 0: K{k,k+1}, 1: K{k+2,k+3}
    const int l    = lane & 15;
    const int tm   = blockIdx.y;
    const int tn   = blockIdx.x;
    const int m    = tm * 16 + l;          // A row for this lane
    const int n    = tn * 16 + l;          // B col for this lane

    v8f acc = {};
    for (int k = 0; k < K; k += 4) {
        const int kk = k + half * 2;
        v2f a, b;
        a.x = A[(size_t)m * K + kk];
        a.y = A[(size_t)m * K + kk + 1];
        if (TRANSB) {
            b.x = B[(size_t)n * K + kk];
            b.y = B[(size_t)n * K + kk + 1];
        } else {
            b.x = B[(size_t)kk * N + n];
            b.y = B[(size_t)(kk + 1) * N + n];
        }
        acc = __builtin_amdgcn_wmma_f32_16x16x4_f32(
            /*neg_a=*/false, a, /*neg_b=*/false, b,
            /*c_mod=*/(short)0, acc, /*reuse_a=*/false, /*reuse_b=*/false);
    }

    // C/D layout: VGPR v -> row (v + 8*half), col = lane&15
    const int row0 = tm * 16 + half * 8;
    const int col  = tn * 16 + l;
    const float bv = BIAS ? bias[col] : 0.0f;
#pragma unroll
    for (int v = 0; v < 8; ++v) {
        C[(size_t)(row0 + v) * N + col] = acc[v] + bv;
    }
}

// ---------------------------------------------------------------------------
// Kernel 3: c[s] = dot(b_i, t[s])   (one wave per segment)
// ---------------------------------------------------------------------------
__global__ void k_cvec(const float* __restrict__ t,
                       const float* __restrict__ b_i,
                       float* __restrict__ cv) {
    const int s = blockIdx.x;
    const int lane = threadIdx.x;
    const float4* tr = (const float4*)(t + (size_t)s * N_H);
    const float4* br = (const float4*)b_i;
    float4 a0 = tr[lane],      w0 = br[lane];
    float4 a1 = tr[lane + 32], w1 = br[lane + 32];
    float acc = a0.x * w0.x + a0.y * w0.y + a0.z * w0.z + a0.w * w0.w
              + a1.x * w1.x + a1.y * w1.y + a1.z * w1.z + a1.w * w1.w;
#pragma unroll
    for (int off = 16; off > 0; off >>= 1) acc += __shfl_xor(acc, off, 32);
    if (lane == 0) cv[s] = acc;
}

// ---------------------------------------------------------------------------
// Kernel 4: logits.  One wave per sample:
//   out[g] = dot(embedding[idx_g], u[seg_g]) + c[seg_g] + b_k
// Lane i loads float4 at element offsets 4*i and 128+4*i -> two contiguous
// 512B bursts per wave.  wave32 xor-tree reduction.
// ---------------------------------------------------------------------------
__global__ void k_logits(const float* __restrict__ emb,
                         const int* __restrict__ pos,
                         const int* __restrict__ neg,
                         const float* __restrict__ u,
                         const float* __restrict__ cv,
                         const float* __restrict__ bk,
                         float* __restrict__ out) {
    const int lane = threadIdx.x & 31;
    const int wave = threadIdx.x >> 5;
    const int g = blockIdx.x * 8 + wave;   // sample id, < 786432

    int idx, seg;
    if (g < N_POS) {
        idx = pos[g];
        seg = g >> 7;                      // /128
    } else {
        const int mneg = g - N_POS;
        idx = neg[mneg];
        seg = mneg / NEG_PER_SEG;          // /640
    }

    const float4* xr = (const float4*)(emb + (size_t)idx * N_H);
    const float4* ur = (const float4*)(u + (size_t)seg * N_H);
    float4 a0 = xr[lane],      b0 = ur[lane];
    float4 a1 = xr[lane + 32], b1 = ur[lane + 32];
    float acc = a0.x * b0.x + a0.y * b0.y + a0.z * b0.z + a0.w * b0.w
              + a1.x * b1.x + a1.y * b1.y + a1.z * b1.z + a1.w * b1.w;
#pragma unroll
    for (int off = 16; off > 0; off >>= 1) acc += __shfl_xor(acc, off, 32);
    if (lane == 0) out[g] = acc + cv[seg] + bk[0];
}

// ---------------------------------------------------------------------------
// Launch: algebraically collapsed pipeline.
//   xbar  = segment_mean(embedding[pos])                    [1024,256]
//   grid  = xbar @ W_i^T + b_i                              (WMMA)
//   t     = grid @ W_k[0]^T                                 (WMMA)
//   u     = t @ W_i                                         (WMMA)
//   c[s]  = b_i . t[s]
//   out   = embedding[samples] . u[seg] + c[seg] + b_k
// ---------------------------------------------------------------------------
extern "C" void kernel_launch(void* const* d_in, const int* in_sizes, int n_in,
                              void* d_out, int out_size, void* d_ws, size_t ws_size,
                              hipStream_t stream) {
    const float* emb = (const float*)d_in[0];
    const int*   gsz = (const int*)d_in[1];
    const int*   pos = (const int*)d_in[2];
    const int*   neg = (const int*)d_in[3];
    const float* W_i = (const float*)d_in[4];
    const float* b_i = (const float*)d_in[5];
    const float* W_k = (const float*)d_in[6];   // [1,256,256] flat
    const float* b_k = (const float*)d_in[7];
    float* out = (float*)d_out;

    float* ws   = (float*)d_ws;
    const size_t MAT = (size_t)N_SEG * N_H;     // 262144 floats
    float* xbar = ws;
    float* grid = ws + 1 * MAT;
    float* t    = ws + 2 * MAT;
    float* u    = ws + 3 * MAT;
    float* cv   = ws + 4 * MAT;                 // 1024 floats

    k_segmean<<<N_SEG, N_H, 0, stream>>>(emb, pos, gsz, xbar);

    dim3 gg(N_H / 16, N_SEG / 16);              // (16, 64) tiles
    k_gemm_wmma<1, 1><<<gg, 32, 0, stream>>>(xbar, W_i, b_i, grid, N_SEG, N_H, N_H);
    k_gemm_wmma<1, 0><<<gg, 32, 0, stream>>>(grid, W_k, nullptr, t, N_SEG, N_H, N_H);
    k_gemm_wmma<0, 0><<<gg, 32, 0, stream>>>(t, W_i, nullptr, u, N_SEG, N_H, N_H);

    k_cvec<<<N_SEG, 32, 0, stream>>>(t, b_i, cv);

    k_logits<<<(N_POS + N_NEG) / 8, 256, 0, stream>>>(emb, pos, neg, u, cv, b_k, out);
}